// AbstractMultilayerClustering_86878598463982
// MI455X (gfx1250) — compile-verified
//
#include <hip/hip_runtime.h>
#include <hip/hip_bf16.h>

typedef float v2f __attribute__((ext_vector_type(2)));
typedef float v4f __attribute__((ext_vector_type(4)));
typedef float v8f __attribute__((ext_vector_type(8)));

#define NOUT  64
#define NPER  5
#define DD1   16
#define DD2   32
#define DTOT  48

__global__ __launch_bounds__(256) void mlc_wmma_kernel(
    const float* __restrict__ x,
    const float* __restrict__ oc,     // (64,16)
    const float* __restrict__ ic,     // (64,5,32)
    long long*   __restrict__ out,
    int n, int numTiles)
{
    __shared__ float s_inner[NOUT * NPER * DD2];   // 10240 f = 40 KB
    __shared__ float s_hcsq_in[NOUT * NPER];       // 0.5*|c|^2 inner
    __shared__ float s_hcsq_out[NOUT];             // 0.5*|c|^2 outer

    const int tid  = threadIdx.x;
    const int lane = tid & 31;
    const int wave = tid >> 5;
    const int h    = lane >> 4;     // lane half (0 or 1)
    const int l15  = lane & 15;

    // ---- block init: inner centers + half-norms into LDS ----
    {
        const v4f* src = (const v4f*)ic;
        v4f*       dst = (v4f*)s_inner;
        for (int i = tid; i < (NOUT * NPER * DD2) / 4; i += 256) dst[i] = src[i];
        for (int j = tid; j < NOUT * NPER; j += 256) {
            const float* c = ic + j * DD2;
            float s = 0.f;
            #pragma unroll
            for (int d = 0; d < DD2; ++d) s = fmaf(c[d], c[d], s);
            s_hcsq_in[j] = 0.5f * s;
        }
        if (tid < NOUT) {
            const float* c = oc + tid * DD1;
            float s = 0.f;
            #pragma unroll
            for (int d = 0; d < DD1; ++d) s = fmaf(c[d], c[d], s);
            s_hcsq_out[tid] = 0.5f * s;
        }
    }
    __syncthreads();

    // ---- per-wave persistent B matrix: outer_centers^T (16x64), f32 ----
    // B tile (4x16) per WMMA step: VGPR j, lane half h -> K = j + 2h.
    v2f B[4][4];
    #pragma unroll
    for (int nt = 0; nt < 4; ++nt) {
        const int ncol = nt * 16 + l15;
        #pragma unroll
        for (int kk = 0; kk < 4; ++kk)
            B[nt][kk] = *(const v2f*)(oc + ncol * DD1 + kk * 4 + 2 * h);
    }
    float cn[4];
    #pragma unroll
    for (int nt = 0; nt < 4; ++nt) cn[nt] = s_hcsq_out[nt * 16 + l15];

    // candidate center indices for this lane (one per N-tile), hoisted
    const int i0 = l15, i1 = 16 + l15, i2 = 32 + l15, i3 = 48 + l15;
    // lane that holds the outer index of point l15 after the r-loop:
    const int oSrcLane = 16 * (l15 >> 3) + (l15 & 7);

    const int waveGlobal = blockIdx.x * 8 + wave;
    const int totalWaves = gridDim.x * 8;

    for (int tile = waveGlobal; tile < numTiles; tile += totalWaves) {
        // ---- A matrix: 16 points x 16 dims, 16x4-f32 layout, 4 K-steps ----
        int row = tile * 16 + l15;
        if (row >= n) row = n - 1;
        const float* xr = x + (size_t)(unsigned)row * DTOT;
        v2f A[4];
        #pragma unroll
        for (int kk = 0; kk < 4; ++kk)
            A[kk] = __builtin_nontemporal_load((const v2f*)(xr + kk * 4 + 2 * h));

        // ---- 16x wmma_f32_16x16x4_f32: dots[16 pts][64 centers] ----
        v8f acc[4];
        #pragma unroll
        for (int nt = 0; nt < 4; ++nt) {
            v8f c = {0.f, 0.f, 0.f, 0.f, 0.f, 0.f, 0.f, 0.f};
            #pragma unroll
            for (int kk = 0; kk < 4; ++kk)
                c = __builtin_amdgcn_wmma_f32_16x16x4_f32(
                        false, A[kk], false, B[nt][kk], (short)0, c, false, false);
            acc[nt] = c;
        }

        // ---- argmin over 64 centers: key = |c|^2/2 - dot  (branchless) ----
        // acc[nt][r] on this lane = dot(point r+8h, center nt*16 + l15)
        int myO = 0;
        #pragma unroll
        for (int r = 0; r < 8; ++r) {
            // per-lane combine over the 4 N-tiles (straight-line selects)
            float v0 = cn[0] - acc[0][r];
            float v1 = cn[1] - acc[1][r];
            float v2 = cn[2] - acc[2][r];
            float v3 = cn[3] - acc[3][r];
            float bv = fminf(fminf(v0, v1), fminf(v2, v3));
            int bi = i3;
            bi = (bv == v2) ? i2 : bi;
            bi = (bv == v1) ? i1 : bi;
            bi = (bv == v0) ? i0 : bi;
            // value-only min butterfly across the 16-lane half
            float mv = bv;
            #pragma unroll
            for (int m = 1; m < 16; m <<= 1)
                mv = fminf(mv, __shfl_xor(mv, m, 32));
            // winning lane (lowest on ties) -> fetch its center index
            unsigned long long bm = __ballot(bv == mv);
            unsigned hm  = (unsigned)(bm >> (16 * h)) & 0xFFFFu;
            int src      = 16 * h + __builtin_ctz(hm);
            int bi_r     = __shfl(bi, src, 32);       // uniform within half
            myO = ((l15 & 7) == r) ? bi_r : myO;      // keep own row's result
        }
        // route outer index of point l15 to both of its stage-2 lanes
        const int o = __shfl(myO, oSrcLane, 32);

        // ---- stage 2: point l15 split across lane pair (h = dim half) ----
        int p  = tile * 16 + l15;
        int pc = (p >= n) ? (n - 1) : p;
        const float* x2 = x + (size_t)(unsigned)pc * DTOT + DD1 + h * 16;
        v4f xv[4];
        #pragma unroll
        for (int q = 0; q < 4; ++q)
            xv[q] = __builtin_nontemporal_load((const v4f*)(x2 + 4 * q));

        float part[NPER];
        #pragma unroll
        for (int k = 0; k < NPER; ++k) {
            const v4f* c4 = (const v4f*)(s_inner + (o * NPER + k) * DD2 + h * 16);
            v2f d2 = {0.f, 0.f};
            #pragma unroll
            for (int q = 0; q < 4; ++q) {
                v4f cv  = c4[q];
                v2f xlo = __builtin_shufflevector(xv[q], xv[q], 0, 1);
                v2f xhi = __builtin_shufflevector(xv[q], xv[q], 2, 3);
                v2f clo = __builtin_shufflevector(cv, cv, 0, 1);
                v2f chi = __builtin_shufflevector(cv, cv, 2, 3);
                d2 = xlo * clo + d2;   // v_pk_fma_f32
                d2 = xhi * chi + d2;   // v_pk_fma_f32
            }
            part[k] = d2.x + d2.y;
        }
        float bv2 = 3.4e38f;
        int   bk  = 0;
        #pragma unroll
        for (int k = 0; k < NPER; ++k) {
            float dot = part[k] + __shfl_xor(part[k], 16, 32);  // combine halves
            float v   = s_hcsq_in[o * NPER + k] - dot;
            bool  t   = v < bv2;
            bk  = t ? k : bk;
            bv2 = t ? v : bv2;
        }
        if (lane < 16 && p < n)
            __builtin_nontemporal_store((long long)(o * NPER + bk), out + p);
    }
}

extern "C" void kernel_launch(void* const* d_in, const int* in_sizes, int n_in,
                              void* d_out, int out_size, void* d_ws, size_t ws_size,
                              hipStream_t stream) {
    const float* x  = (const float*)d_in[0];
    const float* oc = (const float*)d_in[1];
    const float* ic = (const float*)d_in[2];
    long long*   out = (long long*)d_out;

    const int n        = in_sizes[0] / DTOT;        // 1,000,000
    const int numTiles = (n + 15) / 16;             // 62,500 (exact)
    const int blocks   = (numTiles + 7) / 8;        // 8 waves per block

    mlc_wmma_kernel<<<blocks, 256, 0, stream>>>(x, oc, ic, out, n, numTiles);
}